// ReverseLevinsonDurbin_15358803051106
// MI455X (gfx1250) — compile-verified
//
#include <hip/hip_runtime.h>

// Reverse Levinson-Durbin, algebraically collapsed.
//
// Reference: builds unit strictly-upper U (col j = flipped order-j step-down
// poly), V = U^-1, r_j = sum_k V[k,0] E_k V[k,j].
// Since U is unit upper triangular, V[:,0] = e0, so
//     r_j = E0 * x_j,  where U^T x = e0 (forward substitution), x_0 = 1,
//     x_j = -sum_{i<j} x_i * a^(j)_{j-i},
//     E0  = K^2 * prod_m 1/(1 - k_m^2).
// One wave32 lane per (b,t) sample; ~2K flops/sample, all in registers.
// I/O staged through LDS via gfx1250 async global<->LDS b128 transfers
// (ASYNCcnt path), with a plain b128 vmem+ds fallback.

#define NP 25      // M + 1
#define MM 24      // LPC order
#define BLOCK 256  // 8 wave32's per block
#define CH (BLOCK * NP / 4)                // 1600 float4 chunks per block
#define FR (CH / BLOCK)                    // 6 full rounds of 256 float4s
#define TAILCNT (CH - FR * BLOCK)          // 64 float4s in last round

#if defined(__gfx1250__) &&                                        \
    __has_builtin(__builtin_amdgcn_global_load_async_to_lds_b128) && \
    __has_builtin(__builtin_amdgcn_global_store_async_from_lds_b128) && \
    __has_builtin(__builtin_amdgcn_s_wait_asynccnt)
#define HAS_ASYNC_LDS 1
typedef int v4i_async __attribute__((vector_size(16)));
typedef __attribute__((address_space(1))) v4i_async* gptr128_t;  // global int4*
typedef __attribute__((address_space(3))) v4i_async* lptr128_t;  // LDS int4*
#define TO_AS1(p) ((gptr128_t)(uintptr_t)(p))
#define TO_AS3(p) ((lptr128_t)(unsigned int)(uintptr_t)(p))
#else
#define HAS_ASYNC_LDS 0
#endif

__global__ __launch_bounds__(BLOCK)
void rlevdur_kernel(const float* __restrict__ a, float* __restrict__ r, int nprob) {
    __shared__ __align__(16) float lds[BLOCK * NP];  // 25.6 KB staging

    const int tid       = threadIdx.x;
    const int blockBase = blockIdx.x * BLOCK;
    const long long gbase = (long long)blockBase * NP;
    const long long total = (long long)nprob * NP;
    const bool fullBlock = (blockBase + BLOCK) <= nprob;  // block-uniform

    const int tailIdx = FR * BLOCK + (tid & (TAILCNT - 1)); // always in-bounds

    // ---- global -> LDS stage ----
    if (fullBlock) {
        const float4* a4 = (const float4*)(a + gbase);
        float4* lds4 = (float4*)lds;
#if HAS_ASYNC_LDS
#pragma unroll
        for (int rr = 0; rr < FR; ++rr)
            __builtin_amdgcn_global_load_async_to_lds_b128(
                TO_AS1(a4 + rr * BLOCK + tid), TO_AS3(lds4 + rr * BLOCK + tid), 0, 0);
        if (tid < TAILCNT)
            __builtin_amdgcn_global_load_async_to_lds_b128(
                TO_AS1(a4 + tailIdx), TO_AS3(lds4 + tailIdx), 0, 0);
        __builtin_amdgcn_s_wait_asynccnt(0);
#else
        float4 b0 = a4[0 * BLOCK + tid];
        float4 b1 = a4[1 * BLOCK + tid];
        float4 b2 = a4[2 * BLOCK + tid];
        float4 b3 = a4[3 * BLOCK + tid];
        float4 b4 = a4[4 * BLOCK + tid];
        float4 b5 = a4[5 * BLOCK + tid];
        float4 b6 = a4[tailIdx];               // clamped lane, unconditional
        lds4[0 * BLOCK + tid] = b0;
        lds4[1 * BLOCK + tid] = b1;
        lds4[2 * BLOCK + tid] = b2;
        lds4[3 * BLOCK + tid] = b3;
        lds4[4 * BLOCK + tid] = b4;
        lds4[5 * BLOCK + tid] = b5;
        if (tid < TAILCNT) lds4[tailIdx] = b6;
#endif
    } else {
#pragma unroll
        for (int i = 0; i < NP; ++i) {
            const int idx = i * BLOCK + tid;
            if (gbase + idx < total) lds[idx] = a[gbase + idx];
        }
    }
    __syncthreads();

    const int  prob   = blockBase + tid;
    const bool active = prob < nprob;
    const int  lbase  = tid * NP;

    if (active) {
        // a = [K, c1..c24]; working array w = flip([1,c1..c24]) = [c24..c1, 1]
        const float K = lds[lbase + 0];
        float w[NP];
#pragma unroll
        for (int i = 0; i < MM; ++i) w[i] = lds[lbase + (MM - i)];
        w[MM] = 1.0f;

        // ---- phase 1: step-down; extract reflection coeffs, accumulate E0 ----
        // stage m: w[] = flipped a^(24-m) poly; u0 = k_{24-m}
        float kref[MM];
        float E = K * K;
#pragma unroll
        for (int m = 0; m < MM; ++m) {
            const int   L  = MM - 1 - m;           // new working length
            const float u0 = w[0];
            kref[m] = u0;                          // k_{24-m}
            const float t = 1.0f / (1.0f - u0 * u0);
            E *= t;
            float tmp[NP];
#pragma unroll
            for (int i = 0; i < L; ++i) tmp[i] = (w[1 + i] - u0 * w[L - i]) * t;
#pragma unroll
            for (int i = 0; i < L; ++i) w[i] = tmp[i];
        }

        // ---- phase 2: fused step-up + forward substitution ----
        // p[1..j] = a^(j); x_j = -sum_{i=1..j} x_{j-i} * p[i]
        float p[NP];
        float v[NP];
        v[0] = 1.0f;
        p[1] = kref[MM - 1];                       // k_1
        v[1] = -p[1];
#pragma unroll
        for (int j = 2; j <= MM; ++j) {
            const float kj = kref[MM - j];         // k_j
            float tmp[NP];
#pragma unroll
            for (int i = 1; i < j; ++i) tmp[i] = p[i] + kj * p[j - i];
#pragma unroll
            for (int i = 1; i < j; ++i) p[i] = tmp[i];
            p[j] = kj;
            float s = kj;                          // i=j term: v[0]*p[j]
#pragma unroll
            for (int i = 1; i < j; ++i) s += v[j - i] * p[i];
            v[j] = -s;
        }

        // r_j = E0 * x_j -> own LDS region (bank-conflict-free: gcd(25,64)=1)
#pragma unroll
        for (int i = 0; i < NP; ++i) lds[lbase + i] = E * v[i];
    }
    __syncthreads();

    // ---- LDS -> global stage ----
    if (fullBlock) {
        float4* r4 = (float4*)(r + gbase);
        const float4* lds4 = (const float4*)lds;
#if HAS_ASYNC_LDS
#pragma unroll
        for (int rr = 0; rr < FR; ++rr)
            __builtin_amdgcn_global_store_async_from_lds_b128(
                TO_AS1(r4 + rr * BLOCK + tid), TO_AS3(lds4 + rr * BLOCK + tid), 0, 0);
        if (tid < TAILCNT)
            __builtin_amdgcn_global_store_async_from_lds_b128(
                TO_AS1(r4 + tailIdx), TO_AS3(lds4 + tailIdx), 0, 0);
        __builtin_amdgcn_s_wait_asynccnt(0);  // S_ENDPGM would wait-idle anyway
#else
        float4 c0 = lds4[0 * BLOCK + tid];
        float4 c1 = lds4[1 * BLOCK + tid];
        float4 c2 = lds4[2 * BLOCK + tid];
        float4 c3 = lds4[3 * BLOCK + tid];
        float4 c4 = lds4[4 * BLOCK + tid];
        float4 c5 = lds4[5 * BLOCK + tid];
        float4 c6 = lds4[tailIdx];             // clamped lane, unconditional
        r4[0 * BLOCK + tid] = c0;
        r4[1 * BLOCK + tid] = c1;
        r4[2 * BLOCK + tid] = c2;
        r4[3 * BLOCK + tid] = c3;
        r4[4 * BLOCK + tid] = c4;
        r4[5 * BLOCK + tid] = c5;
        if (tid < TAILCNT) r4[tailIdx] = c6;
#endif
    } else {
#pragma unroll
        for (int i = 0; i < NP; ++i) {
            const int idx = i * BLOCK + tid;
            if (gbase + idx < total) r[gbase + idx] = lds[idx];
        }
    }
}

extern "C" void kernel_launch(void* const* d_in, const int* in_sizes, int n_in,
                              void* d_out, int out_size, void* d_ws, size_t ws_size,
                              hipStream_t stream) {
    const float* a = (const float*)d_in[0];
    // d_in[1] = eye: unused after the algebraic collapse (V[:,0] = e0).
    float* r = (float*)d_out;

    const int nprob = in_sizes[0] / NP;            // B*T = 131072
    const int grid  = (nprob + BLOCK - 1) / BLOCK; // 512 blocks of 8 waves
    rlevdur_kernel<<<grid, BLOCK, 0, stream>>>(a, r, nprob);
}